// ProgWalkTokEmbedWithVal_11287174054008
// MI455X (gfx1250) — compile-verified
//
#include <hip/hip_runtime.h>
#include <hip/hip_bf16.h>
#include <stdint.h>

// ---------------------------------------------------------------------------
// ProgWalkTokEmbedWithVal for MI455X (gfx1250)
//
// Memory-bound: 394 MB of output writes dominates (~17us floor @ 23.3 TB/s).
// All embedding tables (~10.4 MB) are L2-resident (192 MB L2).
// FLOPs are negligible (0.54 GFLOP) -> no WMMA; instead we exercise the
// CDNA5 async-memory path (global_load_async_to_lds_b128, ASYNCcnt
// double-buffering via partial s_wait_asynccnt) for the sparse val-gather,
// and non-temporal stores for the output stream so the streaming writes
// don't evict the L2-resident tables.
// ---------------------------------------------------------------------------

typedef float v4f __attribute__((ext_vector_type(4)));
typedef int   v4i __attribute__((ext_vector_type(4)));

#define D_EMBED   256
#define NBATCH    8192
#define L_NODE    16
#define L_EDGE    15
#define SEQ_TOT   (L_NODE + L_EDGE + L_NODE)   // 47
#define ROWS_TOTAL (SEQ_TOT * NBATCH)          // 385024
#define WAVES_PER_BLOCK 4
#define ROWS_PER_WAVE   16
#define THREADS   (WAVES_PER_BLOCK * 32)       // 128

static_assert(ROWS_TOTAL % (WAVES_PER_BLOCK * ROWS_PER_WAVE) == 0, "grid must divide rows");

// --- sinusoidal PE table: pe[l][d], l in [0,16), d in [0,256) --------------
__global__ void pe_init_kernel(float* __restrict__ pe) {
    int i = blockIdx.x * blockDim.x + threadIdx.x;
    if (i >= L_NODE * D_EMBED) return;
    int l = i >> 8;
    int d = i & 255;
    int even = d & ~1;                         // 2*(d/2)
    // div = exp(even * (-ln(10000)/D))
    float div = expf((float)even * (-9.210340371976184f / (float)D_EMBED));
    float ang = (float)l * div;
    pe[i] = (d & 1) ? cosf(ang) : sinf(ang);
}

// --- main fused kernel: each wave32 handles 16 output rows --------------------------
__global__ __launch_bounds__(THREADS) void embed_kernel(
    const int*   __restrict__ node_idx,   // [16, 8192]
    const int*   __restrict__ edge_idx,   // [15, 8192]
    const int*   __restrict__ val_idx,    // [131072, 8]
    const float* __restrict__ val_w,      // [131072, 8]
    const float* __restrict__ node_tab,   // [128, 256]
    const float* __restrict__ edge_tab,   // [32, 256]
    const float* __restrict__ val_tab,    // [10000, 256]
    const float* __restrict__ pe,         // [16, 256]
    float*       __restrict__ out)        // [47, 8192, 256]
{
    // Per-wave staging area for async gathers of val_tab rows: 8 rows x 1KB.
    __shared__ float stage[WAVES_PER_BLOCK][8][D_EMBED];   // 32 KB

    const int lane = threadIdx.x & 31;
    const int wave = threadIdx.x >> 5;
    // LDS byte offset of this wave's staging buffer (generic ptr low 32 bits
    // == workgroup-relative LDS address per aperture layout).
    const unsigned lds_base =
        (unsigned)(uintptr_t)(&stage[wave][0][0]);

    const int wid  = blockIdx.x * WAVES_PER_BLOCK + wave;
    const int row0 = wid * ROWS_PER_WAVE;

    for (int it = 0; it < ROWS_PER_WAVE; ++it) {
        const int row = row0 + it;            // row = seq*8192 + b
        const int seq = row >> 13;

        v4f r0, r1;
        int l;

        if (seq < L_NODE) {
            // ---- node embedding gather (L2-resident table) ----
            l = seq;
            const int t = node_idx[row];
            const v4f* p = (const v4f*)(node_tab + ((size_t)t << 8) + lane * 8);
            r0 = p[0]; r1 = p[1];
        } else if (seq < L_NODE + L_EDGE) {
            // ---- edge embedding gather ----
            l = seq - L_NODE;
            const int t = edge_idx[row - L_NODE * NBATCH];
            const v4f* p = (const v4f*)(edge_tab + ((size_t)t << 8) + lane * 8);
            r0 = p[0]; r1 = p[1];
        } else {
            // ---- sparse weighted gather-reduce (gnn_spmm) ----
            l = seq - (L_NODE + L_EDGE);
            const int n = row - (L_NODE + L_EDGE) * NBATCH;   // = l*8192 + b
            const v4i* ip = (const v4i*)(val_idx + ((size_t)n << 3));
            const v4f* wp = (const v4f*)(val_w   + ((size_t)n << 3));
            const v4i i0 = ip[0], i1 = ip[1];
            const v4f w0 = wp[0], w1 = wp[1];
            const int   rr[8] = { i0.x, i0.y, i0.z, i0.w, i1.x, i1.y, i1.z, i1.w };
            const float ww[8] = { w0.x, w0.y, w0.z, w0.w, w1.x, w1.y, w1.z, w1.w };

            // Staging buffer is reused across row iterations: make sure the
            // previous row's ds reads retired before the async engine
            // overwrites the buffer.
            asm volatile("s_wait_dscnt 0" ::: "memory");

            // Issue all 8 row-gathers (16 async b128 transfers, ASYNCcnt).
            // lane i moves bytes [32i, 32i+32) of val_tab row rr[k] into the
            // k-th 1KB staging slot.
            #pragma unroll
            for (int k = 0; k < 8; ++k) {
                const uint64_t src =
                    (uint64_t)(uintptr_t)(val_tab + ((size_t)rr[k] << 8) + lane * 8);
                const unsigned dst = lds_base + (unsigned)(k * 1024 + lane * 32);
                asm volatile("global_load_async_to_lds_b128 %0, %1, off"
                             :: "v"(dst), "v"(src) : "memory");
                asm volatile("global_load_async_to_lds_b128 %0, %1, off offset:16"
                             :: "v"(dst), "v"(src) : "memory");
            }

            v4f a0 = { 0.f, 0.f, 0.f, 0.f };
            v4f a1 = { 0.f, 0.f, 0.f, 0.f };

            // Async loads complete in order: waiting for ASYNCcnt <= 8 means
            // the first 8 transfers (k = 0..3) have landed, while k = 4..7
            // are still in flight -> consume/transfer overlap.
            asm volatile("s_wait_asynccnt 8" ::: "memory");
            #pragma unroll
            for (int k = 0; k < 4; ++k) {
                const v4f* sp = (const v4f*)(&stage[wave][k][lane * 8]);
                const v4f s0 = sp[0];
                const v4f s1 = sp[1];
                a0 += ww[k] * s0;
                a1 += ww[k] * s1;
            }
            asm volatile("s_wait_asynccnt 0" ::: "memory");
            #pragma unroll
            for (int k = 4; k < 8; ++k) {
                const v4f* sp = (const v4f*)(&stage[wave][k][lane * 8]);
                const v4f s0 = sp[0];
                const v4f s1 = sp[1];
                a0 += ww[k] * s0;
                a1 += ww[k] * s1;
            }
            r0 = a0; r1 = a1;
        }

        // ---- add positional encoding (L2-resident table) ----
        const v4f* pp = (const v4f*)(pe + (l << 8) + lane * 8);
        r0 += pp[0];
        r1 += pp[1];

        // ---- streaming non-temporal store (don't pollute L2) ----
        v4f* op = (v4f*)(out + ((size_t)row << 8) + lane * 8);
        __builtin_nontemporal_store(r0, op);
        __builtin_nontemporal_store(r1, op + 1);
    }
}

extern "C" void kernel_launch(void* const* d_in, const int* in_sizes, int n_in,
                              void* d_out, int out_size, void* d_ws, size_t ws_size,
                              hipStream_t stream) {
    const int*   node_idx = (const int*)  d_in[0];
    const int*   edge_idx = (const int*)  d_in[1];
    const int*   val_idx  = (const int*)  d_in[2];
    const float* val_w    = (const float*)d_in[3];
    const float* node_tab = (const float*)d_in[4];
    const float* edge_tab = (const float*)d_in[5];
    const float* val_tab  = (const float*)d_in[6];
    float* out = (float*)d_out;
    float* pe  = (float*)d_ws;            // 16*256 floats = 16 KB of scratch

    pe_init_kernel<<<(L_NODE * D_EMBED + 255) / 256, 256, 0, stream>>>(pe);

    const int blocks = ROWS_TOTAL / (WAVES_PER_BLOCK * ROWS_PER_WAVE); // 6016
    embed_kernel<<<blocks, THREADS, 0, stream>>>(
        node_idx, edge_idx, val_idx, val_w, node_tab, edge_tab, val_tab, pe, out);
}